// GAT_5342939316766
// MI455X (gfx1250) — compile-verified
//
#include <hip/hip_runtime.h>
#include <hip/hip_bf16.h>

// GAT (2-layer, PyG-style) for MI455X / gfx1250, wave32.
// GEMMs use V_WMMA_F32_16X16X4_F32 (full fp32 precision, CDNA5 matrix pipe).
// Softmax segment-max via monotone float<->uint atomicMax; segment sums via f32 atomics.

#define N_NODES 50000
#define N_EDGES 800000
#define IN_C    128
#define HID     256
#define HEADS   4
#define C1      64
#define OUT_C   40
#define NEG_SLOPE 0.2f

typedef float v8f __attribute__((ext_vector_type(8)));
typedef float v2f __attribute__((ext_vector_type(2)));

// ---- monotone float <-> uint ordering keys (for atomicMax on signed floats) ----
__device__ __forceinline__ unsigned f2key(float f) {
    unsigned b = __float_as_uint(f);
    return (b & 0x80000000u) ? ~b : (b | 0x80000000u);
}
__device__ __forceinline__ float key2f(unsigned k) {
    unsigned b = (k & 0x80000000u) ? (k ^ 0x80000000u) : ~k;
    return __uint_as_float(b);
}
__device__ __forceinline__ float lrelu(float x) { return x > 0.f ? x : NEG_SLOPE * x; }

// ---------------------------------------------------------------------------
// fill kernels
// ---------------------------------------------------------------------------
__global__ void fill_f32(float* __restrict__ p, float v, size_t n) {
    size_t i = (size_t)blockIdx.x * blockDim.x + threadIdx.x;
    if (i < n) p[i] = v;
}
__global__ void fill_u32(unsigned* __restrict__ p, unsigned v, size_t n) {
    size_t i = (size_t)blockIdx.x * blockDim.x + threadIdx.x;
    if (i < n) p[i] = v;
}

// ---------------------------------------------------------------------------
// C[M x ldc] = A[M x K] @ W[Nreal x K]^T   via V_WMMA_F32_16X16X4_F32
// One wave32 per 16x16 output tile. M must be a multiple of 16 (50000 = 3125*16).
// For Nreal not a multiple of 16 (layer 2: 40), out-of-range lanes load a CLAMPED
// W row (branch-free; D column n depends only on B column n, so garbage columns
// are simply never stored). Inner loop: 2x global_load_b64 + 1x v_wmma per K-step.
// ---------------------------------------------------------------------------
__global__ void wmma_gemm_f32(const float* __restrict__ A,
                              const float* __restrict__ W,
                              float* __restrict__ C,
                              int M, int K, int Nreal, int Ntiles, int ldc) {
    int wave = (int)((blockIdx.x * (size_t)blockDim.x + threadIdx.x) >> 5);
    int lane = threadIdx.x & 31;
    int Mtiles = M >> 4;
    if (wave >= Mtiles * Ntiles) return;           // whole-wave exit: EXEC stays all-1s
    int tm = wave % Mtiles;
    int tn = wave / Mtiles;
    int l16  = lane & 15;
    int half = lane >> 4;                          // 0: lanes 0-15, 1: lanes 16-31
    int row  = tm * 16 + l16;                      // A-matrix M index for this lane
    int col  = tn * 16 + l16;                      // B/C/D N index for this lane
    int colc = col < Nreal ? col : (Nreal - 1);    // clamp for loads (branch-free)

    // Pre-bias by the half-wave K split: lanes 0-15 take K = k,k+1; 16-31 take k+2,k+3.
    const float* arow = A + (size_t)row  * K + 2 * half;
    const float* wrow = W + (size_t)colc * K + 2 * half;

    v8f acc = {};
#pragma unroll 4
    for (int k = 0; k < K; k += 4) {
        v2f a = *(const v2f*)(arow + k);           // A 16x4 fragment (2 VGPRs)
        v2f b = *(const v2f*)(wrow + k);           // B 4x16 fragment (B[kk][n] = W[n][kk])
        acc = __builtin_amdgcn_wmma_f32_16x16x4_f32(
            /*neg_a=*/false, a, /*neg_b=*/false, b,
            /*c_mod=*/(short)0, acc, /*reuse_a=*/false, /*reuse_b=*/false);
    }
    if (col < Nreal) {
#pragma unroll
        for (int r = 0; r < 8; ++r) {
            int m = tm * 16 + r + 8 * half;        // C/D layout: VGPR r -> M = r + 8*half
            C[(size_t)m * ldc + col] = acc[r];
        }
    }
}

// ---------------------------------------------------------------------------
// a_src[n,h] = <h[n,h,:], att_src[h,:]> ; same for a_dst.
// ---------------------------------------------------------------------------
__global__ void att_dots(const float* __restrict__ h,
                         const float* __restrict__ att_s,
                         const float* __restrict__ att_d,
                         int n_nodes, int H, int C,
                         float* __restrict__ as_, float* __restrict__ ad_) {
    int t = blockIdx.x * blockDim.x + threadIdx.x;
    if (t >= n_nodes * H) return;
    int n = t / H, hh = t % H;
    const float* row = h + (size_t)n * H * C + (size_t)hh * C;
    float s = 0.f, d = 0.f;
    for (int c = 0; c < C; ++c) {
        float v = row[c];
        s += v * att_s[hh * C + c];
        d += v * att_d[hh * C + c];
    }
    as_[t] = s;
    ad_[t] = d;
}

__device__ __forceinline__ void edge_sd(const long long* __restrict__ ei, int e,
                                        int& s, int& d) {
    if (e < N_EDGES) { s = (int)ei[e]; d = (int)ei[N_EDGES + e]; }
    else             { s = d = e - N_EDGES; }       // self-loops appended
}

// Pass 1: segment max of leaky-relu logits per (dst, head), via uint atomicMax.
__global__ void edge_max(const long long* __restrict__ ei, int H,
                         const float* __restrict__ asrc, const float* __restrict__ adst,
                         unsigned* __restrict__ mkey) {
    int e = blockIdx.x * blockDim.x + threadIdx.x;
    if (e >= N_EDGES + N_NODES) return;
    int s, d; edge_sd(ei, e, s, d);
    for (int h = 0; h < H; ++h) {
        float l = lrelu(asrc[s * H + h] + adst[d * H + h]);
        atomicMax(&mkey[d * H + h], f2key(l));
    }
}

// Pass 2: denom[dst,h] += exp(logit - m[dst,h])
__global__ void edge_expsum(const long long* __restrict__ ei, int H,
                            const float* __restrict__ asrc, const float* __restrict__ adst,
                            const unsigned* __restrict__ mkey, float* __restrict__ denom) {
    int e = blockIdx.x * blockDim.x + threadIdx.x;
    if (e >= N_EDGES + N_NODES) return;
    int s, d; edge_sd(ei, e, s, d);
    for (int h = 0; h < H; ++h) {
        float l = lrelu(asrc[s * H + h] + adst[d * H + h]);
        atomicAdd(&denom[d * H + h], __expf(l - key2f(mkey[d * H + h])));
    }
}

// Pass 3 (layer 1): out[dst, h*64+c] += h1[src, h*64+c] * alpha. One wave per edge;
// lane covers 8 contiguous channels (head = lane/8), float4-coalesced gathers.
__global__ void edge_aggr1(const long long* __restrict__ ei,
                           const float* __restrict__ h1,
                           const float* __restrict__ asrc, const float* __restrict__ adst,
                           const unsigned* __restrict__ mkey, const float* __restrict__ denom,
                           float* __restrict__ out) {
    size_t t = (size_t)blockIdx.x * blockDim.x + threadIdx.x;
    int e = (int)(t >> 5);
    int lane = (int)(t & 31);
    if (e >= N_EDGES + N_NODES) return;
    int s, d; edge_sd(ei, e, s, d);
    int h = lane >> 3;                              // 4 heads x 8 lanes
    float l = lrelu(asrc[s * HEADS + h] + adst[d * HEADS + h]);
    float alpha = __expf(l - key2f(mkey[d * HEADS + h])) / (denom[d * HEADS + h] + 1e-16f);

    const float4* hs = reinterpret_cast<const float4*>(h1 + (size_t)s * HID) + lane * 2;
    float4 v0 = hs[0], v1 = hs[1];
    float* ob = out + (size_t)d * HID + lane * 8;
    atomicAdd(ob + 0, v0.x * alpha); atomicAdd(ob + 1, v0.y * alpha);
    atomicAdd(ob + 2, v0.z * alpha); atomicAdd(ob + 3, v0.w * alpha);
    atomicAdd(ob + 4, v1.x * alpha); atomicAdd(ob + 5, v1.y * alpha);
    atomicAdd(ob + 6, v1.z * alpha); atomicAdd(ob + 7, v1.w * alpha);
}

// Pass 3 (layer 2): H=1, 40 channels. One wave per edge; lane covers c and c+32.
__global__ void edge_aggr2(const long long* __restrict__ ei,
                           const float* __restrict__ h2,
                           const float* __restrict__ asrc, const float* __restrict__ adst,
                           const unsigned* __restrict__ mkey, const float* __restrict__ denom,
                           float* __restrict__ out) {
    size_t t = (size_t)blockIdx.x * blockDim.x + threadIdx.x;
    int e = (int)(t >> 5);
    int lane = (int)(t & 31);
    if (e >= N_EDGES + N_NODES) return;
    int s, d; edge_sd(ei, e, s, d);
    float l = lrelu(asrc[s] + adst[d]);
    float alpha = __expf(l - key2f(mkey[d])) / (denom[d] + 1e-16f);

    atomicAdd(&out[(size_t)d * OUT_C + lane], h2[(size_t)s * OUT_C + lane] * alpha);
    if (lane < OUT_C - 32)
        atomicAdd(&out[(size_t)d * OUT_C + 32 + lane],
                  h2[(size_t)s * OUT_C + 32 + lane] * alpha);
}

// elu(x + bias) elementwise
__global__ void elu_bias(const float* __restrict__ in, const float* __restrict__ b,
                         float* __restrict__ out, size_t total, int C) {
    size_t i = (size_t)blockIdx.x * blockDim.x + threadIdx.x;
    if (i >= total) return;
    float v = in[i] + b[i % C];
    out[i] = v > 0.f ? v : (__expf(v) - 1.f);
}

__global__ void add_bias(float* __restrict__ out, const float* __restrict__ b,
                         size_t total, int C) {
    size_t i = (size_t)blockIdx.x * blockDim.x + threadIdx.x;
    if (i < total) out[i] += b[i % C];
}

static inline int blocksOf(size_t total, int bs) { return (int)((total + bs - 1) / bs); }

extern "C" void kernel_launch(void* const* d_in, const int* in_sizes, int n_in,
                              void* d_out, int out_size, void* d_ws, size_t ws_size,
                              hipStream_t stream) {
    const float*     x    = (const float*)d_in[0];
    const long long* ei   = (const long long*)d_in[1];   // int64 edge_index [2,E]
    const float*     W1   = (const float*)d_in[2];       // [256,128]
    const float*     as1  = (const float*)d_in[3];       // [4,64]
    const float*     ad1  = (const float*)d_in[4];
    const float*     b1   = (const float*)d_in[5];       // [256]
    const float*     W2   = (const float*)d_in[6];       // [40,256]
    const float*     as2  = (const float*)d_in[7];       // [1,40]
    const float*     ad2  = (const float*)d_in[8];
    const float*     b2   = (const float*)d_in[9];       // [40]
    float* out = (float*)d_out;                          // [50000,40]

    // Workspace layout (~106 MB). Regions reused between layers.
    char* ws = (char*)d_ws;
    size_t off = 0;
    auto carve = [&](size_t bytes) { char* p = ws + off; off += (bytes + 255) & ~(size_t)255; return p; };
    float*    h1    = (float*)   carve((size_t)N_NODES * HID * 4);   // 51.2 MB
    float*    buf   = (float*)   carve((size_t)N_NODES * HID * 4);   // 51.2 MB (out1, then h2)
    float*    a_s   = (float*)   carve((size_t)N_NODES * HEADS * 4); // 0.8 MB
    float*    a_d   = (float*)   carve((size_t)N_NODES * HEADS * 4);
    unsigned* mkey  = (unsigned*)carve((size_t)N_NODES * HEADS * 4);
    float*    denom = (float*)   carve((size_t)N_NODES * HEADS * 4);

    const int BS = 256;
    const size_t Eall = N_EDGES + N_NODES;

    // ---------------- Layer 1 ----------------
    fill_u32<<<blocksOf((size_t)N_NODES * HEADS, BS), BS, 0, stream>>>(mkey, 0u, (size_t)N_NODES * HEADS);
    fill_f32<<<blocksOf((size_t)N_NODES * HEADS, BS), BS, 0, stream>>>(denom, 0.f, (size_t)N_NODES * HEADS);

    // h1 = x @ W1^T : M=50000 (3125 tiles), K=128, N=256 (16 tiles)
    {
        size_t waves = (size_t)(N_NODES / 16) * (HID / 16);
        wmma_gemm_f32<<<blocksOf(waves * 32, BS), BS, 0, stream>>>(x, W1, h1, N_NODES, IN_C, HID, HID / 16, HID);
    }
    att_dots<<<blocksOf((size_t)N_NODES * HEADS, BS), BS, 0, stream>>>(h1, as1, ad1, N_NODES, HEADS, C1, a_s, a_d);
    edge_max   <<<blocksOf(Eall, BS), BS, 0, stream>>>(ei, HEADS, a_s, a_d, mkey);
    edge_expsum<<<blocksOf(Eall, BS), BS, 0, stream>>>(ei, HEADS, a_s, a_d, mkey, denom);
    fill_f32<<<blocksOf((size_t)N_NODES * HID, BS), BS, 0, stream>>>(buf, 0.f, (size_t)N_NODES * HID);
    edge_aggr1 <<<blocksOf(Eall * 32, BS), BS, 0, stream>>>(ei, h1, a_s, a_d, mkey, denom, buf);
    // h1 <- elu(out1 + b1)   (h1 region reused as activated features)
    elu_bias<<<blocksOf((size_t)N_NODES * HID, BS), BS, 0, stream>>>(buf, b1, h1, (size_t)N_NODES * HID, HID);

    // ---------------- Layer 2 ----------------
    fill_u32<<<blocksOf((size_t)N_NODES, BS), BS, 0, stream>>>(mkey, 0u, (size_t)N_NODES);
    fill_f32<<<blocksOf((size_t)N_NODES, BS), BS, 0, stream>>>(denom, 0.f, (size_t)N_NODES);

    // h2 = h1 @ W2^T : K=256, Nreal=40 (3 padded tiles), stored [N,40] in buf
    float* h2 = buf;
    {
        size_t waves = (size_t)(N_NODES / 16) * 3;
        wmma_gemm_f32<<<blocksOf(waves * 32, BS), BS, 0, stream>>>(h1, W2, h2, N_NODES, HID, OUT_C, 3, OUT_C);
    }
    att_dots<<<blocksOf((size_t)N_NODES, BS), BS, 0, stream>>>(h2, as2, ad2, N_NODES, 1, OUT_C, a_s, a_d);
    edge_max   <<<blocksOf(Eall, BS), BS, 0, stream>>>(ei, 1, a_s, a_d, mkey);
    edge_expsum<<<blocksOf(Eall, BS), BS, 0, stream>>>(ei, 1, a_s, a_d, mkey, denom);
    fill_f32<<<blocksOf((size_t)N_NODES * OUT_C, BS), BS, 0, stream>>>(out, 0.f, (size_t)N_NODES * OUT_C);
    edge_aggr2 <<<blocksOf(Eall * 32, BS), BS, 0, stream>>>(ei, h2, a_s, a_d, mkey, denom, out);
    add_bias<<<blocksOf((size_t)N_NODES * OUT_C, BS), BS, 0, stream>>>(out, b2, (size_t)N_NODES * OUT_C, OUT_C);
}